// ColorTransform_20375324852755
// MI455X (gfx1250) — compile-verified
//
#include <hip/hip_runtime.h>

// Problem constants (fixed by the reference: B=16, C=3, H=W=512, degree=3, F=19)
#define HW_    (512 * 512)      // 262144, plane stride
#define HW4_   (HW_ / 4)        // 65536 float4-groups per plane
#define BATCH_ 16
#define P4_    (BATCH_ * HW4_)  // 1,048,576 float4-groups total
#define NF_    19

typedef float v4f __attribute__((ext_vector_type(4)));
typedef int   v4i __attribute__((ext_vector_type(4)));

struct F3 { float r, g, b; };

// ---------------------------------------------------------------------------
// Monomial basis in the exact order produced by poly_feature (degree 3, n=3):
//  0:x0 1:x1 2:x2 | 3:x0^2 4:x0x1 5:x0x2 6:x1^2 7:x1x2 8:x2^2 |
//  9:x0^3 10:x0^2x1 11:x0^2x2 12:x0x1^2 13:x0x1x2 14:x0x2^2
// 15:x1^3 16:x1^2x2 17:x1x2^2 18:x2^3
// ---------------------------------------------------------------------------
__device__ __forceinline__ F3 eval3(float x0, float x1, float x2,
                                    const float (&w)[NF_][3],
                                    const float (&bs)[3]) {
  float m[NF_];
  m[0] = x0;        m[1] = x1;        m[2] = x2;
  m[3] = x0 * x0;   m[4] = x0 * x1;   m[5] = x0 * x2;
  m[6] = x1 * x1;   m[7] = x1 * x2;   m[8] = x2 * x2;
  m[9]  = m[3] * x0;  m[10] = m[3] * x1;  m[11] = m[3] * x2;
  m[12] = x0 * m[6];  m[13] = m[4] * x2;  m[14] = x0 * m[8];
  m[15] = m[6] * x1;  m[16] = m[6] * x2;  m[17] = x1 * m[8];
  m[18] = m[8] * x2;
  float a0 = bs[0], a1 = bs[1], a2 = bs[2];
#pragma unroll
  for (int f = 0; f < NF_; ++f) {
    a0 = fmaf(m[f], w[f][0], a0);
    a1 = fmaf(m[f], w[f][1], a1);
    a2 = fmaf(m[f], w[f][2], a2);
  }
  F3 out; out.r = a0; out.g = a1; out.b = a2;
  return out;
}

// Uniform-address coefficient fetch -> SGPRs (s_load), broadcast to all lanes.
__device__ __forceinline__ void load_coeffs(const float* __restrict__ weight,
                                            const float* __restrict__ bias,
                                            float (&w)[NF_][3], float (&bs)[3]) {
#pragma unroll
  for (int f = 0; f < NF_; ++f)
#pragma unroll
    for (int c = 0; c < 3; ++c) w[f][c] = weight[f * 3 + c];
  bs[0] = bias[0]; bs[1] = bias[1]; bs[2] = bias[2];
}

// ---------------------------------------------------------------------------
// Direct streaming kernel (fallback): 1 thread = 4 pixels, 3x b128 NT loads +
// 3x b128 NT stores. Bandwidth-bound; nontemporal hints keep the 100MB stream
// out of the caches.
// ---------------------------------------------------------------------------
__global__ __launch_bounds__(256) void poly_color_direct(
    const float* __restrict__ x, const float* __restrict__ weight,
    const float* __restrict__ bias, float* __restrict__ out) {
  float w[NF_][3], bs[3];
  load_coeffs(weight, bias, w, bs);

  const int g   = blockIdx.x * 256 + threadIdx.x;   // float4-group id
  const int img = g >> 16;                          // g / HW4_
  const int hw  = (g & (HW4_ - 1)) << 2;
  const size_t base = (size_t)img * 3 * HW_ + (size_t)hw;

  v4f c0 = __builtin_nontemporal_load((const v4f*)(x + base));
  v4f c1 = __builtin_nontemporal_load((const v4f*)(x + base + HW_));
  v4f c2 = __builtin_nontemporal_load((const v4f*)(x + base + 2 * HW_));

  v4f o0, o1, o2;
#pragma unroll
  for (int k = 0; k < 4; ++k) {
    F3 t = eval3(c0[k], c1[k], c2[k], w, bs);
    o0[k] = t.r; o1[k] = t.g; o2[k] = t.b;
  }

  __builtin_nontemporal_store(o0, (v4f*)(out + base));
  __builtin_nontemporal_store(o1, (v4f*)(out + base + HW_));
  __builtin_nontemporal_store(o2, (v4f*)(out + base + 2 * HW_));
}

// ---------------------------------------------------------------------------
// CDNA5 async path: stage the input stream into LDS with
// GLOBAL_LOAD_ASYNC_TO_LDS_B128 (ASYNCcnt), 2-deep software pipeline so the
// next tile is in flight while the current one is computed. Each lane stages
// and reads back only its own 3x16B, so no workgroup barrier is required —
// s_wait_asynccnt alone orders LDS visibility for the issuing lane.
// ---------------------------------------------------------------------------
#if __has_builtin(__builtin_amdgcn_global_load_async_to_lds_b128) && \
    __has_builtin(__builtin_amdgcn_s_wait_asynccnt)
#define POLY_HAVE_ASYNC 1
#define AITER 4  // float4-groups per thread

#define AS1_ __attribute__((address_space(1)))
#define AS3_ __attribute__((address_space(3)))

__device__ __forceinline__ void stage_b128(const float* src, v4f* lds_dst) {
  // Prototype (from hipcc diagnostics): (v4i AS1* src, v4i AS3* dst, imm, imm)
  __builtin_amdgcn_global_load_async_to_lds_b128(
      (AS1_ v4i*)src, (AS3_ v4i*)lds_dst, 0, 0);
}

__global__ __launch_bounds__(256) void poly_color_async(
    const float* __restrict__ x, const float* __restrict__ weight,
    const float* __restrict__ bias, float* __restrict__ out) {
  __shared__ v4f buf[2][3][256];  // 24 KB of the 320 KB WGP LDS

  float w[NF_][3], bs[3];
  load_coeffs(weight, bias, w, bs);

  const int tid   = threadIdx.x;
  const int gbase = blockIdx.x * (AITER * 256) + tid;

  // Prologue: stage iteration 0 into slot 0.
  {
    const int g   = gbase;
    const int img = g >> 16;
    const int hw  = (g & (HW4_ - 1)) << 2;
    const size_t base = (size_t)img * 3 * HW_ + (size_t)hw;
    stage_b128(x + base,            &buf[0][0][tid]);
    stage_b128(x + base + HW_,      &buf[0][1][tid]);
    stage_b128(x + base + 2 * HW_,  &buf[0][2][tid]);
  }

#pragma unroll
  for (int it = 0; it < AITER; ++it) {
    const int nxt = it + 1;
    if (nxt < AITER) {  // keep one tile (3 async ops) in flight
      const int g   = gbase + nxt * 256;
      const int img = g >> 16;
      const int hw  = (g & (HW4_ - 1)) << 2;
      const size_t base = (size_t)img * 3 * HW_ + (size_t)hw;
      const int s = nxt & 1;
      stage_b128(x + base,            &buf[s][0][tid]);
      stage_b128(x + base + HW_,      &buf[s][1][tid]);
      stage_b128(x + base + 2 * HW_,  &buf[s][2][tid]);
      __builtin_amdgcn_s_wait_asynccnt(3);  // iteration `it` has landed
    } else {
      __builtin_amdgcn_s_wait_asynccnt(0);
    }

    const int s = it & 1;
    v4f c0 = buf[s][0][tid];
    v4f c1 = buf[s][1][tid];
    v4f c2 = buf[s][2][tid];

    const int g   = gbase + it * 256;
    const int img = g >> 16;
    const int hw  = (g & (HW4_ - 1)) << 2;
    const size_t base = (size_t)img * 3 * HW_ + (size_t)hw;

    v4f o0, o1, o2;
#pragma unroll
    for (int k = 0; k < 4; ++k) {
      F3 t = eval3(c0[k], c1[k], c2[k], w, bs);
      o0[k] = t.r; o1[k] = t.g; o2[k] = t.b;
    }

    __builtin_nontemporal_store(o0, (v4f*)(out + base));
    __builtin_nontemporal_store(o1, (v4f*)(out + base + HW_));
    __builtin_nontemporal_store(o2, (v4f*)(out + base + 2 * HW_));
  }
}
#endif  // async builtins available

extern "C" void kernel_launch(void* const* d_in, const int* in_sizes, int n_in,
                              void* d_out, int out_size, void* d_ws, size_t ws_size,
                              hipStream_t stream) {
  const float* x      = (const float*)d_in[0];  // [16,3,512,512] f32
  const float* weight = (const float*)d_in[1];  // [19,3] f32
  const float* bias   = (const float*)d_in[2];  // [3] f32
  // d_in[3] = degree (int scalar) — fixed at 3, baked into the kernel.
  float* out = (float*)d_out;                   // [16,3,512,512] f32

#ifdef POLY_HAVE_ASYNC
  poly_color_async<<<P4_ / (256 * AITER), 256, 0, stream>>>(x, weight, bias, out);
#else
  poly_color_direct<<<P4_ / 256, 256, 0, stream>>>(x, weight, bias, out);
#endif
}